// ChebConv_41815801594442
// MI455X (gfx1250) — compile-verified
//
#include <hip/hip_runtime.h>

#define V_NODES 100000
#define E_EDGES 1600000
#define B_BATCH 2
#define FIN     32
#define K_CHEB  4
#define FOUT    32

typedef __attribute__((ext_vector_type(2))) float v2f;
typedef __attribute__((ext_vector_type(8))) float v8f;

// ---------------- init kernels ----------------
__global__ __launch_bounds__(256) void fill_zero_f4(float4* __restrict__ y, int n4) {
    int i = blockIdx.x * blockDim.x + threadIdx.x;
    if (i < n4) y[i] = make_float4(0.f, 0.f, 0.f, 0.f);
}

__global__ __launch_bounds__(256) void neg_copy_f4(float4* __restrict__ y,
                                                   const float4* __restrict__ x, int n4) {
    int i = blockIdx.x * blockDim.x + threadIdx.x;
    if (i < n4) {
        float4 v = x[i];
        y[i] = make_float4(-v.x, -v.y, -v.z, -v.w);
    }
}

// ---------------- SPMM: y[b,row,f] += alpha*vals[e]*x[b,col,f] ----------------
// One lane per (edge, feature). Edge index is wave-uniform: readfirstlane forces
// SGPR so rows/cols/vals become scalar loads; the vector pipe carries only the
// coalesced gather + the fp32 L2 atomic (data is L2-resident: 25.6MB << 192MB).
__global__ __launch_bounds__(256) void spmm_edge_atomic(const int*   __restrict__ rows,
                                                        const int*   __restrict__ cols,
                                                        const float* __restrict__ vals,
                                                        const float* __restrict__ x,
                                                        float*       __restrict__ y,
                                                        float alpha) {
    int t = blockIdx.x * blockDim.x + threadIdx.x;
    int f = t & 31;                                     // feature lane
    int e = __builtin_amdgcn_readfirstlane(t >> 5);     // edge index (wave-uniform)
    if (e >= E_EDGES) return;
    int r = rows[e];
    int c = cols[e];
    float av = alpha * vals[e];
#pragma unroll
    for (int b = 0; b < B_BATCH; ++b) {
        size_t base = (size_t)b * V_NODES * FIN;
        float xv = x[base + (size_t)c * FIN + f];
        unsafeAtomicAdd(&y[base + (size_t)r * FIN + f], av * xv);
    }
}

// ---------------- dense contraction via V_WMMA_F32_16X16X4_F32 ----------------
// GEMM: [200000 x 128] @ [128 x 32], contraction index c = k*32 + f.
// One wave computes a 16x32 output tile. Weight is staged in LDS pre-swizzled
// into WMMA B-fragment order so each fragment is a single ds_load_b64 at an
// immediate offset. The two N-tile accumulators are interleaved so consecutive
// WMMAs are independent.
__global__ __launch_bounds__(256) void cheb_gemm_wmma(const float* __restrict__ x0,
                                                      const float* __restrict__ x1,
                                                      const float* __restrict__ x2,
                                                      const float* __restrict__ x3,
                                                      const float* __restrict__ weight, // [FIN,K,FOUT]
                                                      const float* __restrict__ bias,   // [FOUT]
                                                      float*       __restrict__ out) {  // [B*V, FOUT]
    // Fragment-order weight: ldsWf[(((s*2 + nt)*32) + lane)*2 + j]
    //   = W[c = 4*s + 2*(lane>>4) + j][o = nt*16 + (lane&15)],  c = k*32 + f
    __shared__ float ldsWf[32 * 2 * 32 * 2];   // 16KB

    int tid = threadIdx.x;
    for (int i = tid; i < 32 * 2 * 32 * 2; i += 256) {
        int j    = i & 1;
        int ln   = (i >> 1) & 31;
        int nt   = (i >> 6) & 1;
        int s    = i >> 7;
        int o    = nt * 16 + (ln & 15);
        int cidx = 4 * s + 2 * (ln >> 4) + j;   // contraction index = k*32 + f
        int k    = cidx >> 5;
        int fi   = cidx & 31;
        ldsWf[i] = weight[fi * (K_CHEB * FOUT) + k * FOUT + o];
    }
    __syncthreads();

    const int NTILES = (B_BATCH * V_NODES) / 16;  // 12500 (exact)
    int wave = tid >> 5;
    int lane = tid & 31;
    int tile = blockIdx.x * 8 + wave;
    if (tile >= NTILES) return;                   // uniform per wave -> EXEC stays all-1s

    int mbase = tile * 16;
    int mrow  = mbase + (lane & 15);              // A: M = lane%16
    int koff  = (lane >> 4) * 2;                  // A: lanes 0-15 hold K0/K1, 16-31 hold K2/K3

    const float* xs[4] = { x0, x1, x2, x3 };

    // Load all 32 A-fragments (one per k-step): per lane a float2 = global_load_b64.
    v2f a[32];
#pragma unroll
    for (int t = 0; t < 4; ++t) {
        const float* xp = xs[t] + (size_t)mrow * FIN + koff;
#pragma unroll
        for (int s = 0; s < 8; ++s) {
            a[t * 8 + s] = *(const v2f*)(xp + s * 4);
        }
    }

    // Per-lane fragment pointer: byte offset = lane*8 + (s*2+nt)*256 (immediates).
    const v2f* wf = (const v2f*)(ldsWf + lane * 2);

    v8f acc0 = {0.f, 0.f, 0.f, 0.f, 0.f, 0.f, 0.f, 0.f};
    v8f acc1 = {0.f, 0.f, 0.f, 0.f, 0.f, 0.f, 0.f, 0.f};
#pragma unroll
    for (int s = 0; s < 32; ++s) {
        v2f b0 = wf[(s * 2 + 0) * 32];            // ds_load_b64, imm offset
        v2f b1 = wf[(s * 2 + 1) * 32];            // ds_load_b64, imm offset
        acc0 = __builtin_amdgcn_wmma_f32_16x16x4_f32(false, a[s], false, b0,
                                                     (short)0, acc0, false, false);
        acc1 = __builtin_amdgcn_wmma_f32_16x16x4_f32(false, a[s], false, b1,
                                                     (short)0, acc1, false, false);
    }

    // D layout: VGPR i -> row mbase+i (lanes 0-15) / mbase+8+i (lanes 16-31), N = lane%16.
    int ocol = lane & 15;
    float bv0 = bias[ocol];
    float bv1 = bias[16 + ocol];
#pragma unroll
    for (int i = 0; i < 8; ++i) {
        int row = mbase + i + (lane >> 4) * 8;
        out[(size_t)row * FOUT + ocol]      = acc0[i] + bv0;
        out[(size_t)row * FOUT + 16 + ocol] = acc1[i] + bv1;
    }
}

extern "C" void kernel_launch(void* const* d_in, const int* in_sizes, int n_in,
                              void* d_out, int out_size, void* d_ws, size_t ws_size,
                              hipStream_t stream) {
    const float* x0     = (const float*)d_in[0];   // inputs  [B,V,FIN]
    const int*   rows   = (const int*)  d_in[1];   // [E]
    const int*   cols   = (const int*)  d_in[2];   // [E]
    const float* vals   = (const float*)d_in[3];   // [E]
    const float* weight = (const float*)d_in[4];   // [FIN,K,FOUT]
    const float* bias   = (const float*)d_in[5];   // [FOUT]
    float*       out    = (float*)d_out;           // [B,V,FOUT]

    const size_t nterm = (size_t)B_BATCH * V_NODES * FIN;  // 6.4M floats per term
    float* x1 = (float*)d_ws;
    float* x2 = x1 + nterm;
    float* x3 = x2 + nterm;

    const int n4 = (int)(nterm / 4);
    dim3 blk(256);
    dim3 grid_fill((n4 + 255) / 256);
    dim3 grid_spmm((E_EDGES * 32 + 255) / 256);

    // x1 = L @ x0
    hipLaunchKernelGGL(fill_zero_f4, grid_fill, blk, 0, stream, (float4*)x1, n4);
    hipLaunchKernelGGL(spmm_edge_atomic, grid_spmm, blk, 0, stream,
                       rows, cols, vals, x0, x1, 1.0f);
    // x2 = 2*L@x1 - x0
    hipLaunchKernelGGL(neg_copy_f4, grid_fill, blk, 0, stream,
                       (float4*)x2, (const float4*)x0, n4);
    hipLaunchKernelGGL(spmm_edge_atomic, grid_spmm, blk, 0, stream,
                       rows, cols, vals, x1, x2, 2.0f);
    // x3 = 2*L@x2 - x1
    hipLaunchKernelGGL(neg_copy_f4, grid_fill, blk, 0, stream,
                       (float4*)x3, (const float4*)x1, n4);
    hipLaunchKernelGGL(spmm_edge_atomic, grid_spmm, blk, 0, stream,
                       rows, cols, vals, x2, x3, 2.0f);

    // out = stack(x0..x3) @ W + bias   (WMMA fp32)
    const int ntiles  = (B_BATCH * V_NODES) / 16;  // 12500
    dim3 grid_gemm((ntiles + 7) / 8);
    hipLaunchKernelGGL(cheb_gemm_wmma, grid_gemm, blk, 0, stream,
                       x0, x1, x2, x3, weight, bias, out);
}